// DNC_42838003811059
// MI455X (gfx1250) — compile-verified
//
#include <hip/hip_runtime.h>
#include <hip/hip_bf16.h>
#include <math.h>

typedef float v2f __attribute__((ext_vector_type(2)));
typedef float v8f __attribute__((ext_vector_type(8)));

namespace {
constexpr int B_ = 32, T_ = 64, H_ = 256, R_ = 4, M_ = 128, W_ = 256;
constexpr int INP_   = (R_ + 1) * W_;                 // 1280
constexpr int IFACE_ = W_ * R_ + 3 * W_ + 5 * R_ + 3; // 1815
constexpr int OUT_   = H_ + IFACE_;                   // 2071
constexpr int RW_    = R_ * W_;                       // 1024
constexpr int HS_    = 2080;    // padded h row stride (16B-aligned rows)
constexpr int CH_    = 256;     // K-chunk staged in LDS
constexpr int PADF_  = 260;     // LDS row stride (floats): 16B aligned, bank-spread
constexpr int WAVES_ = 4;       // waves per workgroup sharing the A tile
constexpr float DELTA_ = 1e-6f;
// xi sub-vector offsets (post-slice layout of the interface vector)
constexpr int XO_RKEY  = 0;     // R*W  tanh
constexpr int XO_RBETA = 1024;  // R    softplus
constexpr int XO_WKEY  = 1028;  // W    tanh
constexpr int XO_WBETA = 1284;  // 1    softplus
constexpr int XO_ERASE = 1285;  // W    sigmoid
constexpr int XO_WVEC  = 1541;  // W    tanh
constexpr int XO_FREE  = 1797;  // R    sigmoid
constexpr int XO_AG    = 1801;  // 1    sigmoid
constexpr int XO_WG    = 1802;  // 1    sigmoid
constexpr int XO_MODES = 1803;  // 3R   softmax over groups of 3
} // namespace

static __device__ __forceinline__ float sigmoidf_(float x) {
  return 1.0f / (1.0f + __expf(-x));
}
static __device__ __forceinline__ float softplusf_(float x) {
  return x > 20.0f ? x : log1pf(__expf(x));
}

// ---------------------------------------------------------------------------
// Fill helper (state init; deterministic every launch)
// ---------------------------------------------------------------------------
__global__ void k_fill(float* __restrict__ p, int n, float v) {
  int i = blockIdx.x * blockDim.x + threadIdx.x;
  if (i < n) p[i] = v;
}

// ---------------------------------------------------------------------------
// Controller GEMM:  hnew = tanh( x_t @ W_ih^T + hprev @ W_hh^T + b_ih + b_hh )
// 4 waves per workgroup, each owning one 16-wide N-tile; the 32x3351 A block
// ([x_t | hprev]) is staged through LDS in 32x256 chunks and shared by all 4
// waves (4x less L2 activation traffic).  Each wave computes BOTH 16-row
// batch tiles so every weight fragment is loaded once and reused for two
// WMMAs.  fp32 WMMA 16x16x4.
// A-layout (16x4 f32):  lane L: M = L&15, VGPR0 -> K = 2*(L>>4), VGPR1 -> K+1
// B-layout (4x16 f32):  lane L: N = L&15, VGPR0 -> K = 2*(L>>4), VGPR1 -> K+1
// C/D 16x16 f32: VGPR v: lanes 0-15 -> M=v, lanes 16-31 -> M=v+8, N = L&15
// ---------------------------------------------------------------------------
__global__ void __launch_bounds__(WAVES_ * 32)
k_gemm_h(const float* __restrict__ x,
         const float* __restrict__ Wih,
         const float* __restrict__ Whh,
         const float* __restrict__ bih,
         const float* __restrict__ bhh,
         const float* __restrict__ hprev,
         float* __restrict__ hnew, int t) {
  __shared__ float sA[32 * PADF_];          // 33,280 B
  const int tid  = threadIdx.x;
  const int lane = tid & 31;
  const int wave = tid >> 5;
  const int lo = lane & 15, hi = lane >> 4;
  const int tile  = blockIdx.x * WAVES_ + wave;
  const int ncol0 = tile * 16 + lo;
  const bool nok  = ncol0 < OUT_;
  const int ncol  = nok ? ncol0 : (OUT_ - 1);

  const float* wihrow = Wih + (size_t)ncol * INP_;
  const float* whhrow = Whh + (size_t)ncol * OUT_;

  const float bias = nok ? (bih[ncol0] + bhh[ncol0]) : 0.0f;
  v8f c0, c1;
#pragma unroll
  for (int v = 0; v < 8; ++v) { c0[v] = bias; c1[v] = bias; }

  // ---- Region 1: K in [0,1280), A = x_t (rows 16B-aligned) ----
  for (int cb = 0; cb < INP_; cb += CH_) {
    for (int q = tid; q < 32 * (CH_ / 4); q += WAVES_ * 32) {
      const int row = q >> 6, quad = q & 63;
      const float4 src =
          *(const float4*)(x + ((size_t)row * T_ + t) * INP_ + cb + quad * 4);
      *(float4*)(sA + row * PADF_ + quad * 4) = src;
    }
    __syncthreads();
    const float* wp = wihrow + cb;
    for (int k0 = 0; k0 < CH_; k0 += 4) {
      const int kc = k0 + 2 * hi;
      v2f bm = { wp[kc], wp[kc + 1] };
      v2f a0 = *(const v2f*)(sA + lo * PADF_ + kc);
      v2f a1 = *(const v2f*)(sA + (lo + 16) * PADF_ + kc);
      c0 = __builtin_amdgcn_wmma_f32_16x16x4_f32(false, a0, false, bm,
                                                 (short)0, c0, false, false);
      c1 = __builtin_amdgcn_wmma_f32_16x16x4_f32(false, a1, false, bm,
                                                 (short)0, c1, false, false);
    }
    __syncthreads();
  }

  // ---- Region 2: K_rel in [0,2048), A = hprev (padded stride HS_) ----
  for (int cb = 0; cb < 2048; cb += CH_) {
    for (int q = tid; q < 32 * (CH_ / 4); q += WAVES_ * 32) {
      const int row = q >> 6, quad = q & 63;
      const float4 src =
          *(const float4*)(hprev + (size_t)row * HS_ + cb + quad * 4);
      *(float4*)(sA + row * PADF_ + quad * 4) = src;
    }
    __syncthreads();
    const float* wp = whhrow + cb;
    for (int k0 = 0; k0 < CH_; k0 += 4) {
      const int kc = k0 + 2 * hi;
      v2f bm = { wp[kc], wp[kc + 1] };
      v2f a0 = *(const v2f*)(sA + lo * PADF_ + kc);
      v2f a1 = *(const v2f*)(sA + (lo + 16) * PADF_ + kc);
      c0 = __builtin_amdgcn_wmma_f32_16x16x4_f32(false, a0, false, bm,
                                                 (short)0, c0, false, false);
      c1 = __builtin_amdgcn_wmma_f32_16x16x4_f32(false, a1, false, bm,
                                                 (short)0, c1, false, false);
    }
    __syncthreads();
  }

  // ---- Tail: K_rel in [2048, 2071); LDS zero-padded to 24, weights guarded ----
  {
    for (int q = tid; q < 32 * 24; q += WAVES_ * 32) {
      const int row = q / 24, col = q % 24;
      const int k = 2048 + col;
      sA[row * PADF_ + col] =
          (k < OUT_) ? hprev[(size_t)row * HS_ + k] : 0.0f;
    }
    __syncthreads();
    const float* wp = whhrow + 2048;
    for (int k0 = 0; k0 < 24; k0 += 4) {
      const int kc = k0 + 2 * hi;
      const float b0v = (2048 + kc     < OUT_) ? wp[kc]     : 0.0f;
      const float b1v = (2048 + kc + 1 < OUT_) ? wp[kc + 1] : 0.0f;
      v2f bm = { b0v, b1v };
      v2f a0 = *(const v2f*)(sA + lo * PADF_ + kc);
      v2f a1 = *(const v2f*)(sA + (lo + 16) * PADF_ + kc);
      c0 = __builtin_amdgcn_wmma_f32_16x16x4_f32(false, a0, false, bm,
                                                 (short)0, c0, false, false);
      c1 = __builtin_amdgcn_wmma_f32_16x16x4_f32(false, a1, false, bm,
                                                 (short)0, c1, false, false);
    }
  }

  if (nok) {
#pragma unroll
    for (int v = 0; v < 8; ++v) {
      const int b0 = v + 8 * hi;        // batch rows 0..15 / 16..31
      hnew[(size_t)b0 * HS_ + ncol0]        = tanhf(c0[v]);
      hnew[(size_t)(b0 + 16) * HS_ + ncol0] = tanhf(c1[v]);
    }
  }
}

// ---------------------------------------------------------------------------
// Interface activations: xi = act(h[:, H:])
// ---------------------------------------------------------------------------
__global__ void k_xi(const float* __restrict__ h, float* __restrict__ xi) {
  const int b = blockIdx.x;
  const float* hb = h + (size_t)b * HS_ + H_;
  float* xb = xi + (size_t)b * IFACE_;
  for (int j = threadIdx.x; j < IFACE_; j += blockDim.x) {
    const float v = hb[j];
    float r;
    if (j < XO_RBETA)        r = tanhf(v);
    else if (j < XO_WKEY)    r = softplusf_(v);
    else if (j < XO_WBETA)   r = tanhf(v);
    else if (j < XO_ERASE)   r = softplusf_(v);
    else if (j < XO_WVEC)    r = sigmoidf_(v);
    else if (j < XO_FREE)    r = tanhf(v);
    else if (j < XO_MODES)   r = sigmoidf_(v);   // free, alloc_g, write_g
    else {
      const int base = XO_MODES + ((j - XO_MODES) / 3) * 3;
      const float a0 = hb[base], a1 = hb[base + 1], a2 = hb[base + 2];
      const float mx = fmaxf(a0, fmaxf(a1, a2));
      const float s = __expf(a0 - mx) + __expf(a1 - mx) + __expf(a2 - mx);
      r = __expf(v - mx) / s;
    }
    xb[j] = r;
  }
}

// ---------------------------------------------------------------------------
// Usage update (uses carried ww / rw from previous step)
// ---------------------------------------------------------------------------
__global__ void k_usage(const float* __restrict__ xi, const float* __restrict__ ww,
                        const float* __restrict__ rw, float* __restrict__ usage) {
  const int b = blockIdx.x, m = threadIdx.x;
  const float* xb = xi + (size_t)b * IFACE_;
  float u = usage[b * M_ + m];
  u = u + (1.0f - u) * ww[b * M_ + m];
  float psi = 1.0f;
#pragma unroll
  for (int r = 0; r < R_; ++r)
    psi *= (1.0f - xb[XO_FREE + r] * rw[((size_t)b * R_ + r) * M_ + m]);
  u *= psi;
  u = DELTA_ + (1.0f - DELTA_) * u;
  usage[b * M_ + m] = u;
}

// ---------------------------------------------------------------------------
// Per-batch ascending stable sort of usage (bitonic, idx tiebreak) + allocation
// ---------------------------------------------------------------------------
__global__ void k_sort_alloc(const float* __restrict__ usage,
                             float* __restrict__ alloc) {
  __shared__ float val[M_];
  __shared__ int idx[M_];
  const int b = blockIdx.x, tid = threadIdx.x;
  val[tid] = usage[b * M_ + tid];
  idx[tid] = tid;
  __syncthreads();
  for (int k = 2; k <= M_; k <<= 1) {
    for (int j = k >> 1; j > 0; j >>= 1) {
      const int ixj = tid ^ j;
      if (ixj > tid) {
        const bool up = ((tid & k) == 0);
        const float v0 = val[tid], v1 = val[ixj];
        const int i0 = idx[tid], i1 = idx[ixj];
        const bool gt = (v0 > v1) || (v0 == v1 && i0 > i1);
        if (gt == up) { val[tid] = v1; val[ixj] = v0; idx[tid] = i1; idx[ixj] = i0; }
      }
      __syncthreads();
    }
  }
  if (tid == 0) {
    float run = 1.0f;
    for (int j = 0; j < M_; ++j) {
      alloc[b * M_ + idx[j]] = (1.0f - val[j]) * run;
      run *= val[j];
    }
  }
}

// ---------------------------------------------------------------------------
// Write content weighting (softmax over M) fused with ww = g*(ga*alloc+(1-ga)*c)
// ---------------------------------------------------------------------------
__global__ void k_write_w(const float* __restrict__ xi, const float* __restrict__ mem,
                          const float* __restrict__ alloc, float* __restrict__ ww,
                          float* __restrict__ wwsum) {
  __shared__ float skey[W_];
  __shared__ float red[M_];
  __shared__ float s_knorm, s_max, s_sum;
  const int b = blockIdx.x, m = threadIdx.x;
  const float* xb = xi + (size_t)b * IFACE_;
  float ks = 0.0f;
  for (int w = m; w < W_; w += M_) {
    const float kv = xb[XO_WKEY + w];
    skey[w] = kv;
    ks += kv * kv;
  }
  red[m] = ks; __syncthreads();
  for (int s = M_ / 2; s > 0; s >>= 1) { if (m < s) red[m] += red[m + s]; __syncthreads(); }
  if (m == 0) s_knorm = sqrtf(red[0]) + DELTA_;
  __syncthreads();

  const float* mrow = mem + ((size_t)b * M_ + m) * W_;
  float dot = 0.0f, ms = 0.0f;
  for (int w = 0; w < W_; ++w) {
    const float mv = mrow[w];
    dot += skey[w] * mv;
    ms += mv * mv;
  }
  const float beta = xb[XO_WBETA];
  const float sim = beta * dot / (s_knorm * (sqrtf(ms) + DELTA_));

  red[m] = sim; __syncthreads();
  for (int s = M_ / 2; s > 0; s >>= 1) { if (m < s) red[m] = fmaxf(red[m], red[m + s]); __syncthreads(); }
  if (m == 0) s_max = red[0];
  __syncthreads();
  const float e = __expf(sim - s_max);
  red[m] = e; __syncthreads();
  for (int s = M_ / 2; s > 0; s >>= 1) { if (m < s) red[m] += red[m + s]; __syncthreads(); }
  if (m == 0) s_sum = red[0];
  __syncthreads();
  const float wcw = e / s_sum;

  const float ag = xb[XO_AG], wg = xb[XO_WG];
  const float w_ = wg * (ag * alloc[b * M_ + m] + (1.0f - ag) * wcw);
  ww[b * M_ + m] = w_;
  red[m] = w_; __syncthreads();
  for (int s = M_ / 2; s > 0; s >>= 1) { if (m < s) red[m] += red[m + s]; __syncthreads(); }
  if (m == 0) wwsum[b] = red[0];
}

// ---------------------------------------------------------------------------
// Memory write:  mem = mem*(1 - ww⊗erase) + ww⊗write_vec     grid (M, B) x W
// ---------------------------------------------------------------------------
__global__ void k_mem_update(const float* __restrict__ xi, const float* __restrict__ ww,
                             float* __restrict__ mem) {
  const int m = blockIdx.x, b = blockIdx.y, w = threadIdx.x;
  const float* xb = xi + (size_t)b * IFACE_;
  const float wwm = ww[b * M_ + m];
  const size_t o = ((size_t)b * M_ + m) * W_ + w;
  mem[o] = mem[o] * (1.0f - wwm * xb[XO_ERASE + w]) + wwm * xb[XO_WVEC + w];
}

// ---------------------------------------------------------------------------
// Link update (reads OLD prec):  grid (M, B) x M
// ---------------------------------------------------------------------------
__global__ void k_link(const float* __restrict__ ww, const float* __restrict__ prec,
                       float* __restrict__ link) {
  const int i = blockIdx.x, b = blockIdx.y, j = threadIdx.x;
  const float wwi = ww[b * M_ + i], wwj = ww[b * M_ + j];
  const float pj = prec[b * M_ + j];
  const size_t o = ((size_t)b * M_ + i) * M_ + j;
  float l = link[o];
  l = (1.0f - wwi - wwj) * l + wwi * pj;
  link[o] = (i == j) ? 0.0f : l;
}

__global__ void k_prec(const float* __restrict__ ww, const float* __restrict__ wwsum,
                       float* __restrict__ prec) {
  const int b = blockIdx.x, m = threadIdx.x;
  prec[b * M_ + m] = (1.0f - wwsum[b]) * prec[b * M_ + m] + ww[b * M_ + m];
}

// ---------------------------------------------------------------------------
// Forward/backward weights from NEW link, OLD rw:  grid (R, B) x M
// ---------------------------------------------------------------------------
__global__ void k_fwd_bwd(const float* __restrict__ link, const float* __restrict__ rw,
                          float* __restrict__ fwdw, float* __restrict__ bwdw) {
  __shared__ float srw[M_];
  const int r = blockIdx.x, b = blockIdx.y, i = threadIdx.x;
  srw[i] = rw[((size_t)b * R_ + r) * M_ + i];
  __syncthreads();
  const float* lb = link + (size_t)b * M_ * M_;
  float f = 0.0f, bw = 0.0f;
  for (int j = 0; j < M_; ++j) {
    f  += lb[(size_t)i * M_ + j] * srw[j];
    bw += lb[(size_t)j * M_ + i] * srw[j];
  }
  fwdw[((size_t)b * R_ + r) * M_ + i] = f;
  bwdw[((size_t)b * R_ + r) * M_ + i] = bw;
}

// ---------------------------------------------------------------------------
// Read content weighting over NEW mem:  grid (R, B) x M
// ---------------------------------------------------------------------------
__global__ void k_read_cw(const float* __restrict__ xi, const float* __restrict__ mem,
                          float* __restrict__ rcw) {
  __shared__ float skey[W_];
  __shared__ float red[M_];
  __shared__ float s_knorm, s_max, s_sum;
  const int r = blockIdx.x, b = blockIdx.y, m = threadIdx.x;
  const float* xb = xi + (size_t)b * IFACE_;
  float ks = 0.0f;
  for (int w = m; w < W_; w += M_) {
    const float kv = xb[XO_RKEY + r * W_ + w];
    skey[w] = kv;
    ks += kv * kv;
  }
  red[m] = ks; __syncthreads();
  for (int s = M_ / 2; s > 0; s >>= 1) { if (m < s) red[m] += red[m + s]; __syncthreads(); }
  if (m == 0) s_knorm = sqrtf(red[0]) + DELTA_;
  __syncthreads();

  const float* mrow = mem + ((size_t)b * M_ + m) * W_;
  float dot = 0.0f, ms = 0.0f;
  for (int w = 0; w < W_; ++w) {
    const float mv = mrow[w];
    dot += skey[w] * mv;
    ms += mv * mv;
  }
  const float beta = xb[XO_RBETA + r];
  const float sim = beta * dot / (s_knorm * (sqrtf(ms) + DELTA_));

  red[m] = sim; __syncthreads();
  for (int s = M_ / 2; s > 0; s >>= 1) { if (m < s) red[m] = fmaxf(red[m], red[m + s]); __syncthreads(); }
  if (m == 0) s_max = red[0];
  __syncthreads();
  const float e = __expf(sim - s_max);
  red[m] = e; __syncthreads();
  for (int s = M_ / 2; s > 0; s >>= 1) { if (m < s) red[m] += red[m + s]; __syncthreads(); }
  if (m == 0) s_sum = red[0];
  __syncthreads();
  rcw[((size_t)b * R_ + r) * M_ + m] = e / s_sum;
}

// ---------------------------------------------------------------------------
// rw = modes·(bwd, content, fwd);  read_vec = rw @ mem:  grid (R, B) x W
// ---------------------------------------------------------------------------
__global__ void k_rw_readvec(const float* __restrict__ xi, const float* __restrict__ bwdw,
                             const float* __restrict__ rcw, const float* __restrict__ fwdw,
                             const float* __restrict__ mem, float* __restrict__ rw,
                             float* __restrict__ readf) {
  __shared__ float srw[M_];
  const int r = blockIdx.x, b = blockIdx.y, tid = threadIdx.x;
  const float* xb = xi + (size_t)b * IFACE_;
  if (tid < M_) {
    const float m0 = xb[XO_MODES + r * 3 + 0];
    const float m1 = xb[XO_MODES + r * 3 + 1];
    const float m2 = xb[XO_MODES + r * 3 + 2];
    const size_t o = ((size_t)b * R_ + r) * M_ + tid;
    const float v = m0 * bwdw[o] + m1 * rcw[o] + m2 * fwdw[o];
    rw[o] = v;
    srw[tid] = v;
  }
  __syncthreads();
  const float* mb_ = mem + (size_t)b * M_ * W_;
  float acc = 0.0f;
  for (int m = 0; m < M_; ++m) acc += srw[m] * mb_[(size_t)m * W_ + tid];
  readf[(size_t)b * RW_ + r * W_ + tid] = acc;
}

// ---------------------------------------------------------------------------
// Output GEMM: o = [h[:, :H] | read_flat] @ W_out^T + b_out   (fp32 WMMA)
// One wave per 16-wide N-tile; both batch tiles fused for B-fragment reuse.
// ---------------------------------------------------------------------------
__global__ void k_gemm_out(const float* __restrict__ hnew, const float* __restrict__ readf,
                           const float* __restrict__ Wout, const float* __restrict__ bout,
                           float* __restrict__ out, int t) {
  const int lane = threadIdx.x;
  const int lo = lane & 15, hi = lane >> 4;
  const int nb = blockIdx.x * 16;   // output feature (H=256, exact)
  const int ncol = nb + lo;

  const float* hrow0 = hnew  + (size_t)lo * HS_;           // K in [0,256)
  const float* hrow1 = hnew  + (size_t)(lo + 16) * HS_;
  const float* rrow0 = readf + (size_t)lo * RW_;           // K in [256,1280)
  const float* rrow1 = readf + (size_t)(lo + 16) * RW_;
  const float* wrow  = Wout  + (size_t)ncol * INP_;

  const float bias = bout[ncol];
  v8f c0, c1;
#pragma unroll
  for (int v = 0; v < 8; ++v) { c0[v] = bias; c1[v] = bias; }

  {
    const v2f* hp0 = (const v2f*)hrow0;         // HS_ even -> 8B aligned
    const v2f* hp1 = (const v2f*)hrow1;
    const v2f* wp  = (const v2f*)wrow;          // INP_ even -> 8B aligned
    for (int k0 = 0; k0 < H_; k0 += 4) {
      const int kh = (k0 >> 1) + hi;
      v2f bm = wp[kh];
      v2f a0 = hp0[kh];
      v2f a1 = hp1[kh];
      c0 = __builtin_amdgcn_wmma_f32_16x16x4_f32(false, a0, false, bm,
                                                 (short)0, c0, false, false);
      c1 = __builtin_amdgcn_wmma_f32_16x16x4_f32(false, a1, false, bm,
                                                 (short)0, c1, false, false);
    }
  }
  {
    const v2f* rp0 = (const v2f*)rrow0;         // RW_ even -> 8B aligned
    const v2f* rp1 = (const v2f*)rrow1;
    const v2f* wp  = (const v2f*)wrow;
    for (int k0 = H_; k0 < INP_; k0 += 4) {
      const int kh = (k0 >> 1) + hi;
      const int kr = kh - (H_ >> 1);
      v2f bm = wp[kh];
      v2f a0 = rp0[kr];
      v2f a1 = rp1[kr];
      c0 = __builtin_amdgcn_wmma_f32_16x16x4_f32(false, a0, false, bm,
                                                 (short)0, c0, false, false);
      c1 = __builtin_amdgcn_wmma_f32_16x16x4_f32(false, a1, false, bm,
                                                 (short)0, c1, false, false);
    }
  }
#pragma unroll
  for (int v = 0; v < 8; ++v) {
    const int b0 = v + 8 * hi;
    out[((size_t)b0 * T_ + t) * H_ + ncol]        = c0[v];
    out[((size_t)(b0 + 16) * T_ + t) * H_ + ncol] = c1[v];
  }
}

// ---------------------------------------------------------------------------
extern "C" void kernel_launch(void* const* d_in, const int* in_sizes, int n_in,
                              void* d_out, int out_size, void* d_ws, size_t ws_size,
                              hipStream_t stream) {
  (void)in_sizes; (void)n_in; (void)out_size; (void)ws_size;
  const float* x    = (const float*)d_in[0];
  const float* Wih  = (const float*)d_in[1];
  const float* bih  = (const float*)d_in[2];
  const float* Whh  = (const float*)d_in[3];
  const float* bhh  = (const float*)d_in[4];
  const float* Wout = (const float*)d_in[5];
  const float* bout = (const float*)d_in[6];
  float* out = (float*)d_out;

  // Workspace carve-up (floats, 256B-aligned chunks); total ~7.5 MB.
  float* p = (float*)d_ws;
  auto take = [&p](size_t n) { float* q = p; p += ((n + 63) / 64) * 64; return q; };
  float* h0     = take((size_t)B_ * HS_);
  float* h1     = take((size_t)B_ * HS_);
  float* mem    = take((size_t)B_ * M_ * W_);
  float* link   = take((size_t)B_ * M_ * M_);
  float* prec   = take((size_t)B_ * M_);
  float* rw     = take((size_t)B_ * R_ * M_);
  float* ww     = take((size_t)B_ * M_);
  float* usage  = take((size_t)B_ * M_);
  float* readf  = take((size_t)B_ * RW_);
  float* xi     = take((size_t)B_ * IFACE_);
  float* alloc  = take((size_t)B_ * M_);
  float* wwsum  = take((size_t)B_);
  float* fwdw   = take((size_t)B_ * R_ * M_);
  float* bwdw   = take((size_t)B_ * R_ * M_);
  float* rcw    = take((size_t)B_ * R_ * M_);

  auto fill = [&](float* ptr, size_t n, float v) {
    k_fill<<<dim3((unsigned)((n + 255) / 256)), 256, 0, stream>>>(ptr, (int)n, v);
  };
  fill(h0,    (size_t)B_ * HS_,      0.0f);
  fill(mem,   (size_t)B_ * M_ * W_,  DELTA_);
  fill(link,  (size_t)B_ * M_ * M_,  0.0f);
  fill(prec,  (size_t)B_ * M_,       0.0f);
  fill(rw,    (size_t)B_ * R_ * M_,  DELTA_);
  fill(ww,    (size_t)B_ * M_,       DELTA_);
  fill(usage, (size_t)B_ * M_,       0.0f);
  fill(readf, (size_t)B_ * RW_,      0.0f);

  constexpr int NTILES = (OUT_ + 15) / 16;                    // 130
  const dim3 gemmH((NTILES + WAVES_ - 1) / WAVES_);           // 33 workgroups
  const dim3 gemmO(H_ / 16);                                  // 16 N-tiles
  const dim3 gMB(M_, B_);
  const dim3 gRB(R_, B_);

  for (int t = 0; t < T_; ++t) {
    const float* hprev = (t & 1) ? h1 : h0;
    float* hnew        = (t & 1) ? h0 : h1;
    k_gemm_h   <<<gemmH, WAVES_ * 32, 0, stream>>>(x, Wih, Whh, bih, bhh,
                                                   hprev, hnew, t);
    k_xi       <<<B_,   256,  0, stream>>>(hnew, xi);
    k_usage    <<<B_,   M_,   0, stream>>>(xi, ww, rw, usage);
    k_sort_alloc<<<B_,  M_,   0, stream>>>(usage, alloc);
    k_write_w  <<<B_,   M_,   0, stream>>>(xi, mem, alloc, ww, wwsum);
    k_mem_update<<<gMB, W_,   0, stream>>>(xi, ww, mem);
    k_link     <<<gMB,  M_,   0, stream>>>(ww, prec, link);
    k_prec     <<<B_,   M_,   0, stream>>>(ww, wwsum, prec);
    k_fwd_bwd  <<<gRB,  M_,   0, stream>>>(link, rw, fwdw, bwdw);
    k_read_cw  <<<gRB,  M_,   0, stream>>>(xi, mem, rcw);
    k_rw_readvec<<<gRB, W_,   0, stream>>>(xi, bwdw, rcw, fwdw, mem, rw, readf);
    k_gemm_out <<<gemmO, 32,  0, stream>>>(hnew, readf, Wout, bout, out, t);
  }
}